// Decoder_11922829214033
// MI455X (gfx1250) — compile-verified
//
#include <hip/hip_runtime.h>
#include <hip/hip_bf16.h>

#define HIDDEN 128
#define WAVES_PER_BLOCK 8

typedef __attribute__((ext_vector_type(2))) float v2f;
typedef __attribute__((ext_vector_type(4))) float v4f;
typedef __attribute__((ext_vector_type(8))) float v8f;

// out[e] = <eh[e], W_e> + <nh[src[e]], W_s> + <nh[dst[e]], W_d> + b
// One wave32 computes 16 edges via a chain of V_WMMA_F32_16X16X4_F32:
//   A = 16(edges) x 4(K)  feature slice   (f32, 2 VGPRs)
//   B = 4(K) x 16(N)      W slice broadcast across all 16 columns
//   C = 16x16 f32 accumulator; column N=0 carries the real result.
__global__ __launch_bounds__(256) void edge_decoder_wmma(
    const float* __restrict__ nh,    // (nNodes, 128)
    const float* __restrict__ eh,    // (nEdges, 128)
    const int*   __restrict__ eidx,  // (2, nEdges)  row0=src, row1=dst
    const float* __restrict__ W,     // (384,)  [W_e | W_s | W_d]
    const float* __restrict__ b,     // (1,)
    float* __restrict__ out,         // (nEdges,)
    int nEdges, int nGroups)
{
    const int wave = threadIdx.x >> 5;
    const int lane = threadIdx.x & 31;
    const int g = blockIdx.x * WAVES_PER_BLOCK + wave;
    if (g >= nGroups) return;                 // wave-uniform: EXEC stays full

    const int row  = lane & 15;               // edge-in-group (matrix row M)
    const int half = lane >> 4;               // K sub-pair selector
    const int e0   = g << 4;

    int e = e0 + row;
    if (e >= nEdges) e = nEdges - 1;          // tail: clamp loads, guard stores

    const int si = eidx[e];                   // src gather index
    const int di = eidx[nEdges + e];          // dst gather index

    // Per-lane base pointers: lane supplies K = c*4 + 2*half + {0,1}
    const float* ea = eh + (size_t)e  * HIDDEN + 2 * half;
    const float* sa = nh + (size_t)si * HIDDEN + 2 * half;
    const float* da = nh + (size_t)di * HIDDEN + 2 * half;
    const float* we = W              + 2 * half;   // W_e
    const float* ws = we + HIDDEN;                 // W_s
    const float* wd = ws + HIDDEN;                 // W_d

    v8f acc = {0.f, 0.f, 0.f, 0.f, 0.f, 0.f, 0.f, 0.f};

    #pragma unroll 4
    for (int c = 0; c < HIDDEN / 4; ++c) {
        const int off = c * 4;
        // A operands: aligned 8B loads (base 512B-aligned, off*4 16B, half*8)
        v2f aE = *(const v2f*)(ea + off);
        v2f aS = *(const v2f*)(sa + off);
        v2f aD = *(const v2f*)(da + off);
        // B operands: W chunk broadcast over N (hits L0, W is 1.5 KB)
        v2f bE = *(const v2f*)(we + off);
        v2f bS = *(const v2f*)(ws + off);
        v2f bD = *(const v2f*)(wd + off);

        acc = __builtin_amdgcn_wmma_f32_16x16x4_f32(false, aE, false, bE,
                                                    (short)0, acc, false, false);
        acc = __builtin_amdgcn_wmma_f32_16x16x4_f32(false, aS, false, bS,
                                                    (short)0, acc, false, false);
        acc = __builtin_amdgcn_wmma_f32_16x16x4_f32(false, aD, false, bD,
                                                    (short)0, acc, false, false);
    }

    // C/D layout: lane 0 VGPR r -> (M=r,   N=0)
    //             lane16 VGPR r -> (M=8+r, N=0)
    if (row == 0) {
        const float bb    = b[0];
        const int   mbase = half * 8;
        const int   obase = e0 + mbase;
        if (e0 + 16 <= nEdges) {
            // Full group (all but at most the last): two 16B stores per lane.
            v4f lo = { acc[0] + bb, acc[1] + bb, acc[2] + bb, acc[3] + bb };
            v4f hi = { acc[4] + bb, acc[5] + bb, acc[6] + bb, acc[7] + bb };
            *(v4f*)(out + obase)     = lo;   // obase is 16-float aligned or +8
            *(v4f*)(out + obase + 4) = hi;
        } else {
            #pragma unroll
            for (int r = 0; r < 8; ++r) {
                const int o = obase + r;
                if (o < nEdges) out[o] = acc[r] + bb;
            }
        }
    }
}

extern "C" void kernel_launch(void* const* d_in, const int* in_sizes, int n_in,
                              void* d_out, int out_size, void* d_ws, size_t ws_size,
                              hipStream_t stream) {
    const float* nh   = (const float*)d_in[0];   // node_hidden
    const float* eh   = (const float*)d_in[1];   // edge_hidden
    const int*   eidx = (const int*)  d_in[2];   // edge_index (int32 on device)
    const float* W    = (const float*)d_in[3];   // (384,1)
    const float* b    = (const float*)d_in[4];   // (1,)
    float*       out  = (float*)d_out;

    const int nEdges  = in_sizes[1] / HIDDEN;
    const int nGroups = (nEdges + 15) / 16;
    const int blocks  = (nGroups + WAVES_PER_BLOCK - 1) / WAVES_PER_BLOCK;

    edge_decoder_wmma<<<blocks, 256, 0, stream>>>(nh, eh, eidx, W, b, out,
                                                  nEdges, nGroups);
}